// PointPillarsScatter_88313117540620
// MI455X (gfx1250) — compile-verified
//
#include <hip/hip_runtime.h>

// ---------------------------------------------------------------------------
// PointPillarsScatter for MI455X (gfx1250). 0 FLOPs, pure data movement.
// Roofline: ~294MB / 23.3 TB/s ~= 12.6us. WMMA is inapplicable (no matmul);
// the gfx1250 paths that matter are the async LDS-DMA engine, prefetch, and
// NT streaming stores.
//
// Pipeline (main path):
//   1) pp_init_map:    fill int32 map[B][512][512] with -1 (4MB, L2-resident)
//   2) pp_scatter_map: map[b][x][y] = pillar index (unique), + prefetch row
//   3) pp_gather_rows: block per (b,x) row; per 128-cell y-segment:
//        - async-DMA map segment into LDS        (global_load_async_to_lds_b128)
//        - async-DMA 256B feature rows into LDS  (global_load_async_to_lds_b128)
//        - stream [c][y] output slabs with non-temporal stores
//      Feature reads are exactly-once coalesced 256B bursts instead of
//      64 random 4B gathers per pillar; every output byte written once.
// ---------------------------------------------------------------------------

#define PP_C    64
#define PP_NX   512
#define PP_NY   512
#define PP_SEG  128   // y-cells per segment
#define PP_PAD  68    // LDS tile row pitch in floats: 272B = 17*16B, so every
                      // row start is 16B-aligned for B128 DMA writes; output
                      // reads see only a 2-way bank conflict ((4y+c) mod 64).

typedef float __attribute__((ext_vector_type(4))) f32x4;
typedef int   __attribute__((ext_vector_type(4))) i32x4;

__device__ __forceinline__ unsigned lds_off(const void* p) {
  // Flat LDS addresses carry the workgroup-relative LDS byte offset in the
  // low 32 bits (aperture in the high bits) — truncate to get the DS address.
  return (unsigned)(size_t)p;
}

__device__ __forceinline__ void async_load_b128(unsigned lds_addr,
                                                const void* gaddr) {
  asm volatile("global_load_async_to_lds_b128 %0, %1, off"
               :: "v"(lds_addr), "v"(gaddr)
               : "memory");
}

__device__ __forceinline__ void wait_async0() {
  asm volatile("s_wait_asynccnt 0x0" ::: "memory");
}

// ---------------------------------------------------------------------------
// Pass 1: map = -1 everywhere (regular-temporal: keep the 4MB map in L2).
// ---------------------------------------------------------------------------
__global__ __launch_bounds__(256) void pp_init_map(int* __restrict__ map,
                                                   int n_int4) {
  int i = blockIdx.x * 256 + threadIdx.x;
  const int stride = gridDim.x * 256;
  const i32x4 neg1 = {-1, -1, -1, -1};
  for (; i < n_int4; i += stride) {
    reinterpret_cast<i32x4*>(map)[i] = neg1;
  }
}

// ---------------------------------------------------------------------------
// Pass 2: scatter pillar index into map[b][x][y]; prefetch its feature row
// (gfx1250 global_prefetch_b8) so pass-3 DMA reads hit L2.
// ---------------------------------------------------------------------------
__global__ __launch_bounds__(256) void pp_scatter_map(
    const int* __restrict__ coords, const float* __restrict__ feat,
    int* __restrict__ map, int N) {
  int i = blockIdx.x * 256 + threadIdx.x;
  if (i >= N) return;
  const int b = coords[4 * i + 0];
  const int y = coords[4 * i + 2];
  const int x = coords[4 * i + 3];
  map[(((long)b * PP_NX + x) << 9) + y] = i;
  const float* row = feat + (long)i * PP_C;  // 256 bytes
  __builtin_prefetch(row + 0,  0, 2);
  __builtin_prefetch(row + 32, 0, 2);
}

// ---------------------------------------------------------------------------
// Pass 3: block per (b,x) row. LDS: map segment + 16B-aligned feature tile.
// ---------------------------------------------------------------------------
__global__ __launch_bounds__(256) void pp_gather_rows(
    const float* __restrict__ feat, const int* __restrict__ map,
    float* __restrict__ out) {
  __shared__ int   smap[PP_SEG];              // 512 B
  __shared__ float ftile[PP_SEG * PP_PAD];    // 34,816 B

  const int bx  = blockIdx.x;                 // b*NX + x
  const int x   = bx & (PP_NX - 1);
  const int b   = bx >> 9;
  const int tid = threadIdx.x;

  const int* maprow = map + ((long)bx << 9);                  // map[b][x][.]
  float* outrow = out + ((long)b * PP_C << 18) + ((long)x << 9);

  const int row16 = tid >> 4;        // 0..15 : feature-row group
  const int l16   = tid & 15;        // lane within a 256B feature row
  const int y_loc = tid & (PP_SEG - 1);
  const int c_off = tid >> 7;        // 0..1

  for (int seg = 0; seg < 4; ++seg) {
    const int y0 = seg * PP_SEG;

    // --- DMA the 128-entry map segment into LDS (wave 0) -----------------
    if (tid < 32) {
      async_load_b128(lds_off(smap) + tid * 16, maprow + y0 + tid * 4);
      wait_async0();
    }
    __syncthreads();

    // --- DMA 256B feature rows of non-empty cells into the LDS tile ------
    for (int r = row16; r < PP_SEG; r += 16) {
      const int n = smap[r];
      if (n >= 0) {
        // LDS byte addr = (r*68 + l16*4)*4 = r*272 + l16*16 : 16B-aligned.
        async_load_b128(lds_off(ftile + r * PP_PAD + l16 * 4),
                        feat + ((long)n << 6) + l16 * 4);
      }
    }
    wait_async0();
    __syncthreads();

    // --- stream the output: 64 c-planes x 128 y, NT stores ---------------
    const int n_mine = smap[y_loc];
    const float* frow = ftile + y_loc * PP_PAD;
    float* o = outrow + y0 + y_loc + ((long)c_off << 18);
    for (int cb = 0; cb < PP_C; cb += 2) {
      const int c = cb + c_off;
      const float v = (n_mine >= 0) ? frow[c] : 0.0f;
      __builtin_nontemporal_store(v, o + ((long)cb << 18));
    }
    __syncthreads();  // protect smap/ftile before next segment's DMA
  }
}

// ---------------------------------------------------------------------------
// Fallback (workspace too small): zero-fill + direct scatter.
// ---------------------------------------------------------------------------
__global__ __launch_bounds__(256) void pp_zero(float* __restrict__ out,
                                               int n_f4) {
  int i = blockIdx.x * 256 + threadIdx.x;
  const int stride = gridDim.x * 256;
  const f32x4 z = {0.0f, 0.0f, 0.0f, 0.0f};
  for (; i < n_f4; i += stride) {
    __builtin_nontemporal_store(z, reinterpret_cast<f32x4*>(out) + i);
  }
}

__global__ __launch_bounds__(256) void pp_scatter_direct(
    const float* __restrict__ feat, const int* __restrict__ coords,
    float* __restrict__ out, int N) {
  const long t = (long)blockIdx.x * 256 + threadIdx.x;
  const int i = (int)(t >> 6);
  const int c = (int)(t & 63);
  if (i >= N) return;
  const int b = coords[4 * i + 0];
  const int y = coords[4 * i + 2];
  const int x = coords[4 * i + 3];
  out[(((((long)b * PP_C + c) * PP_NX) + x) << 9) + y] =
      feat[((long)i << 6) + c];
}

extern "C" void kernel_launch(void* const* d_in, const int* in_sizes, int n_in,
                              void* d_out, int out_size, void* d_ws,
                              size_t ws_size, hipStream_t stream) {
  const float* feat = (const float*)d_in[0];
  const int* coords = (const int*)d_in[1];
  const int N = in_sizes[1] / 4;                    // number of pillars
  const int B = out_size / (PP_C * PP_NX * PP_NY);  // batch from out size
  float* out = (float*)d_out;

  const size_t map_bytes = (size_t)B * PP_NX * PP_NY * sizeof(int);

  if (d_ws != nullptr && ws_size >= map_bytes) {
    int* map = (int*)d_ws;
    const int n_int4 = (int)(map_bytes / 16);
    pp_init_map<<<(n_int4 + 255) / 256, 256, 0, stream>>>(map, n_int4);
    pp_scatter_map<<<(N + 255) / 256, 256, 0, stream>>>(coords, feat, map, N);
    pp_gather_rows<<<B * PP_NX, 256, 0, stream>>>(feat, map, out);
  } else {
    const int n_f4 = out_size / 4;
    pp_zero<<<4096, 256, 0, stream>>>(out, n_f4);
    const long tot = (long)N * PP_C;
    pp_scatter_direct<<<(int)((tot + 255) / 256), 256, 0, stream>>>(
        feat, coords, out, N);
  }
}